// GraphConvolution_53249004535835
// MI455X (gfx1250) — compile-verified
//
#include <hip/hip_runtime.h>
#include <hip/hip_bf16.h>

typedef __attribute__((ext_vector_type(2))) float v2f;
typedef __attribute__((ext_vector_type(8))) float v8f;

#define K_DIM 512
#define N_DIM 128
#define ROWS_PER_BLOCK 80              // 50000 = 625 * 80 (exact)
#define M_TILES 5                      // 5 x 16 rows per wave
#define LDA 516                        // padded LDS row stride: bank-conflict-free
#define LDS_BYTES (ROWS_PER_BLOCK * LDA * 4)  // ~161 KB of the 320 KB WGP LDS

// ---------------------------------------------------------------------------
// Kernel 1: support = x @ W  via V_WMMA_F32_16X16X4_F32
//   block = 256 threads (8 waves). Block stages an 80-row A panel in LDS
//   (161 KB); wave w owns N columns [16w,16w+16) and 5 M-tiles, so each B
//   fragment (read from L2-resident W) is reused by 5 WMMAs. K loop:
//   512/4 = 128 steps -> 640 v_wmma per wave.
// A frag (ISA 7.12.2, f32 16x4): lanes 0-15 = M rows, K {0,1}; lanes 16-31
// K {2,3}. C/D: VGPR r holds M = r + 8*(lane>>4), N = lane&15.
// ---------------------------------------------------------------------------
__global__ __launch_bounds__(256) void gcn_gemm_wmma(const float* __restrict__ x,
                                                     const float* __restrict__ w,
                                                     float* __restrict__ support,
                                                     int n_rows) {
  extern __shared__ float As[];  // [ROWS_PER_BLOCK][LDA], dynamic (>64KB)

  const int m0  = blockIdx.x * ROWS_PER_BLOCK;
  const int tid = threadIdx.x;

  // Cooperative A-panel load: 80 rows * 128 float4, coalesced; rows clamped.
  for (int i = tid; i < ROWS_PER_BLOCK * (K_DIM / 4); i += 256) {
    int r   = i / (K_DIM / 4);
    int c4  = i % (K_DIM / 4);
    int row = m0 + r;
    row = (row < n_rows) ? row : (n_rows - 1);
    ((float4*)(As + r * LDA))[c4] = ((const float4*)(x + (size_t)row * K_DIM))[c4];
  }
  __syncthreads();

  const int wave  = tid >> 5;
  const int lane  = tid & 31;
  const int mrow  = lane & 15;         // A row within tile for this lane
  const int khalf = (lane >> 4) * 2;   // K sub-offset: 0 (lanes 0-15) / 2 (16-31)
  const int ncol  = wave * 16 + (lane & 15);

  v8f acc[M_TILES] = {};

#pragma unroll 4
  for (int k = 0; k < K_DIM; k += 4) {
    // One B fragment per k-step, reused across 5 M-tiles.
    v2f b;
    b.x = w[(size_t)(k + khalf) * N_DIM + ncol];
    b.y = w[(size_t)(k + khalf + 1) * N_DIM + ncol];
#pragma unroll
    for (int t = 0; t < M_TILES; ++t) {
      v2f a;
      a.x = As[(t * 16 + mrow) * LDA + k + khalf];
      a.y = As[(t * 16 + mrow) * LDA + k + khalf + 1];
      acc[t] = __builtin_amdgcn_wmma_f32_16x16x4_f32(
          false, a, false, b, (short)0, acc[t], false, false);
    }
  }

  // Store per C/D layout. Fast path (always taken for n_rows % 80 == 0).
  if (m0 + ROWS_PER_BLOCK <= n_rows) {
#pragma unroll
    for (int t = 0; t < M_TILES; ++t)
#pragma unroll
      for (int r = 0; r < 8; ++r) {
        int m = m0 + t * 16 + r + 8 * (lane >> 4);
        support[(size_t)m * N_DIM + ncol] = acc[t][r];
      }
  } else {
#pragma unroll
    for (int t = 0; t < M_TILES; ++t)
#pragma unroll
      for (int r = 0; r < 8; ++r) {
        int m = m0 + t * 16 + r + 8 * (lane >> 4);
        if (m < n_rows) support[(size_t)m * N_DIM + ncol] = acc[t][r];
      }
  }
}

// ---------------------------------------------------------------------------
// Kernel 2: COO SpMM scatter-add. Each wave processes edges in batches of 32:
// lane l loads edge (base+l) coalesced, then (r,c,v) are broadcast lane-by-
// lane with wave32 shuffles. For each edge all 32 lanes gather a float4 of
// support[c] and atomically add v*s into out[r]. support/out are ~25.6 MB
// each -> resident in the 192 MB L2, so gathers + atomics run at L2 speed.
// ---------------------------------------------------------------------------
__global__ __launch_bounds__(256) void gcn_spmm_atomic(const long long* __restrict__ rows,
                                                       const long long* __restrict__ cols,
                                                       const float* __restrict__ vals,
                                                       const float* __restrict__ support,
                                                       float* __restrict__ out,
                                                       int n_edges) {
  const int gid   = blockIdx.x * blockDim.x + threadIdx.x;
  const int wid   = gid >> 5;
  const int lane  = gid & 31;
  const int waves = (gridDim.x * blockDim.x) >> 5;

  for (int base = wid * 32; base < n_edges; base += waves * 32) {
    const int e = base + lane;
    int   my_r = 0, my_c = 0;
    float my_v = 0.f;
    if (e < n_edges) {
      my_r = (int)rows[e];   // coalesced: one edge per lane
      my_c = (int)cols[e];
      my_v = vals[e];
    }
    const int cnt = (n_edges - base < 32) ? (n_edges - base) : 32;
    for (int j = 0; j < cnt; ++j) {
      const int   r = __shfl(my_r, j);
      const int   c = __shfl(my_c, j);
      const float v = __shfl(my_v, j);
      const float4 s = *(const float4*)(support + (size_t)c * N_DIM + lane * 4);
      float* o = out + (size_t)r * N_DIM + lane * 4;
      atomicAdd(o + 0, v * s.x);
      atomicAdd(o + 1, v * s.y);
      atomicAdd(o + 2, v * s.z);
      atomicAdd(o + 3, v * s.w);
    }
  }
}

// ---------------------------------------------------------------------------
// Kernel 3: out = relu(out + bias), float4-vectorized in-place epilogue.
// ---------------------------------------------------------------------------
__global__ __launch_bounds__(256) void gcn_bias_relu(float* __restrict__ out,
                                                     const float* __restrict__ bias,
                                                     int n_vec4) {
  const int stride = gridDim.x * blockDim.x;
  for (int i = blockIdx.x * blockDim.x + threadIdx.x; i < n_vec4; i += stride) {
    float4 v = ((float4*)out)[i];
    const float4 b = ((const float4*)bias)[i & ((N_DIM / 4) - 1)];
    v.x = fmaxf(v.x + b.x, 0.f);
    v.y = fmaxf(v.y + b.y, 0.f);
    v.z = fmaxf(v.z + b.z, 0.f);
    v.w = fmaxf(v.w + b.w, 0.f);
    ((float4*)out)[i] = v;
  }
}

// ---------------------------------------------------------------------------
// Launch: GEMM -> zero out -> SpMM atomics -> bias+ReLU, all on `stream`.
// Inputs: [0]=x (N*512 f32), [1]=edge_row (E i64), [2]=edge_col (E i64),
//         [3]=edge_val (E f32), [4]=weight (512*128 f32), [5]=bias (128 f32)
// ---------------------------------------------------------------------------
extern "C" void kernel_launch(void* const* d_in, const int* in_sizes, int n_in,
                              void* d_out, int out_size, void* d_ws, size_t ws_size,
                              hipStream_t stream) {
  const float*     x    = (const float*)d_in[0];
  const long long* erow = (const long long*)d_in[1];
  const long long* ecol = (const long long*)d_in[2];
  const float*     eval = (const float*)d_in[3];
  const float*     wgt  = (const float*)d_in[4];
  const float*     bias = (const float*)d_in[5];
  float*           out  = (float*)d_out;
  float*           supp = (float*)d_ws;  // N*128 f32 = 25.6 MB scratch

  const int n_nodes = in_sizes[0] / K_DIM;  // 50000
  const int n_edges = in_sizes[1];          // 1.6M

  // 1) support = x @ W  (WMMA f32, 80-row panels, B reused 5x)
  const int m_blocks = (n_nodes + ROWS_PER_BLOCK - 1) / ROWS_PER_BLOCK;  // 625
  gcn_gemm_wmma<<<m_blocks, 256, LDS_BYTES, stream>>>(x, wgt, supp, n_nodes);

  // 2) out = 0 (graph-capture-safe async memset)
  hipMemsetAsync(out, 0, (size_t)out_size * sizeof(float), stream);

  // 3) scatter-add messages
  gcn_spmm_atomic<<<4096, 256, 0, stream>>>(erow, ecol, eval, supp, out, n_edges);

  // 4) bias + ReLU
  gcn_bias_relu<<<2048, 256, 0, stream>>>(out, bias, out_size / 4);
}